// DTCWTForward_42949672961265
// MI455X (gfx1250) — compile-verified
//
#include <hip/hip_runtime.h>

#define H1 256
#define W1 256

typedef __attribute__((ext_vector_type(2))) float v2f;
typedef __attribute__((ext_vector_type(8))) float v8f;

#if defined(__HIP_DEVICE_COMPILE__) && defined(__gfx1250__) && __has_builtin(__builtin_amdgcn_wmma_f32_16x16x4_f32)
#define USE_WMMA 1
#else
#define USE_WMMA 0
#endif

// Symmetric ("reflect with edge repeat") index: ...,1,0 | 0..l-1 | l-1,l-2,...
__device__ __forceinline__ int reflect_idx(int x, int l) {
  int p = 2 * l;
  int m = x % p;
  m += (m < 0) ? p : 0;
  return (m >= l) ? (p - 1 - m) : m;
}

__device__ __forceinline__ float bandtap(const float* __restrict__ f, int taps, int i) {
  return (i >= 0 && i < taps) ? f[i] : 0.0f;
}

// ---------------------------------------------------------------------------
// Level 1: fused rowfilter(h0,h1) + colfilter + q2c.
// Grid: (16 row-blocks, 256 images). Block: 128 threads (4 waves).
// Each block: 16 output rows x 256 cols. LDS: xt/lo/hi = 22 rows x 256 cols.
// Column filtering runs as banded 16x16x4 f32 WMMA per 16x16 tile (4 tiles/wave).
// ---------------------------------------------------------------------------
__global__ __launch_bounds__(128) void dtcwt_l1(
    const float* __restrict__ x,
    const float* __restrict__ h0, const float* __restrict__ h1,
    float* __restrict__ ll_out, float* __restrict__ yh_out)
{
  extern __shared__ float smem[];
  float* lo = smem;               // 22*256
  float* hi = lo + 22 * 256;      // 22*256
  float* xt = hi + 22 * 256;      // 22*256 (dead after stage 2; reused as q2c scratch)
  float* sc = xt;

  const int rb = blockIdx.x;
  const int bc = blockIdx.y;
  const int t = threadIdx.x;
  const int lane = t & 31;
  const int wave = t >> 5;

  const float* __restrict__ xin = x + (size_t)bc * (H1 * W1);
  const int r0 = rb * 16 - 3;     // top halo of 3 (max m for h1)

  // ---- Stage 1: global -> LDS (22 rows incl. vertical symmetric halo) ----
  for (int idx = t; idx < 22 * 256; idx += 128) {
    int rr = idx >> 8;
    int cc = idx & 255;
    int gr = reflect_idx(r0 + rr, H1);
    xt[idx] = xin[(size_t)gr * W1 + cc];
  }
  if (rb + 1 < 16) {  // warm L2 for next row block -> global_prefetch_b8
    __builtin_prefetch(xin + (size_t)(rb * 16 + 16) * W1 + t * 4, 0, 1);
  }
  __syncthreads();

  // ---- Stage 2: horizontal filters; thread t owns cols 2t, 2t+1 ----
  for (int rr = 0; rr < 22; rr++) {
    const float* xr = xt + rr * 256;
    #pragma unroll
    for (int u = 0; u < 2; u++) {
      int c0 = 2 * t + u;
      float aLo = 0.0f, aHi = 0.0f;
      #pragma unroll
      for (int k = 0; k < 5; k++) aLo += h0[k] * xr[reflect_idx(c0 + k - 2, W1)];
      #pragma unroll
      for (int k = 0; k < 7; k++) aHi += h1[k] * xr[reflect_idx(c0 + k - 3, W1)];
      lo[rr * 256 + c0] = aLo;
      hi[rr * 256 + c0] = aHi;
    }
  }
  __syncthreads();

  // ---- Stage 3: column filters as banded WMMA + q2c ----
  // A fragment layout (16x4 f32): VGPR0 = {K=0 | K=2}, VGPR1 = {K=1 | K=3},
  // lanes 0-15 hold M=lane, lanes 16-31 hold M=lane-16.
  const int rA = lane & 15;
  const int kHalf = (lane < 16) ? 0 : 2;
  v2f A0[6], A1[6];
  #pragma unroll
  for (int kk = 0; kk < 6; kk++) {
    int kb = 4 * kk + kHalf;
    // ll/hl band: Y[r] = sum_k h0[k] * Rows[r+1+k]  -> A[r][kcol] = h0[kcol-r-1]
    A0[kk].x = bandtap(h0, 5, kb - rA - 1);
    A0[kk].y = bandtap(h0, 5, kb + 1 - rA - 1);
    // lh/hh band: Y[r] = sum_k h1[k] * Rows[r+k]    -> A[r][kcol] = h1[kcol-r]
    A1[kk].x = bandtap(h1, 7, kb - rA);
    A1[kk].y = bandtap(h1, 7, kb + 1 - rA);
  }

  const int TS = 17;  // scratch tile stride (bank-conflict pad)
  float* scw = sc + wave * (3 * 16 * TS);
  const float inv_sqrt2 = 0.70710678118654752440f;

  for (int tt = 0; tt < 4; tt++) {
    const int ct = wave + 4 * tt;       // 16-col tile index 0..15
    const int cb = ct * 16;
    const int col = cb + (lane & 15);

    v8f dll = {}, dlh = {}, dhl = {}, dhh = {};
#if USE_WMMA
    #pragma unroll
    for (int kk = 0; kk < 6; kk++) {
      int kb = 4 * kk + kHalf;
      int ka0 = (kb < 21) ? kb : 21;          // rows >=22 have zero A coeffs; clamp
      int ka1 = (kb + 1 < 21) ? (kb + 1) : 21;
      v2f bL, bH;
      bL.x = lo[ka0 * 256 + col];  bL.y = lo[ka1 * 256 + col];
      bH.x = hi[ka0 * 256 + col];  bH.y = hi[ka1 * 256 + col];
      dll = __builtin_amdgcn_wmma_f32_16x16x4_f32(false, A0[kk], false, bL, (short)0, dll, false, false);
      dlh = __builtin_amdgcn_wmma_f32_16x16x4_f32(false, A1[kk], false, bL, (short)0, dlh, false, false);
      dhl = __builtin_amdgcn_wmma_f32_16x16x4_f32(false, A0[kk], false, bH, (short)0, dhl, false, false);
      dhh = __builtin_amdgcn_wmma_f32_16x16x4_f32(false, A1[kk], false, bH, (short)0, dhh, false, false);
    }
#else
    // Scalar fallback producing the identical D-fragment layout.
    #pragma unroll
    for (int v = 0; v < 8; v++) {
      int M = v + ((lane >> 4) << 3);
      float sll = 0.f, slh = 0.f, shl = 0.f, shh = 0.f;
      #pragma unroll
      for (int k = 0; k < 5; k++) {
        sll += h0[k] * lo[(M + 1 + k) * 256 + col];
        shl += h0[k] * hi[(M + 1 + k) * 256 + col];
      }
      #pragma unroll
      for (int k = 0; k < 7; k++) {
        slh += h1[k] * lo[(M + k) * 256 + col];
        shh += h1[k] * hi[(M + k) * 256 + col];
      }
      dll[v] = sll; dlh[v] = slh; dhl[v] = shl; dhh[v] = shh;
    }
#endif

    // LL (full resolution) straight to workspace from D-fragment layout.
    #pragma unroll
    for (int v = 0; v < 8; v++) {
      int M = v + ((lane >> 4) << 3);
      ll_out[((size_t)bc * H1 + (size_t)(rb * 16 + M)) * W1 + col] = dll[v];
    }

    // Spill highpass fragments to per-wave LDS scratch for the 2x2 recombine.
    #pragma unroll
    for (int v = 0; v < 8; v++) {
      int M = v + ((lane >> 4) << 3);
      int n = lane & 15;
      scw[0 * 16 * TS + M * TS + n] = dlh[v];
      scw[1 * 16 * TS + M * TS + n] = dhl[v];
      scw[2 * 16 * TS + M * TS + n] = dhh[v];
    }
    __syncthreads();

    // q2c + orientation packing: 8x8 2x2-blocks per tile, 2 blocks per lane.
    #pragma unroll
    for (int q = 0; q < 2; q++) {
      int blk = lane + q * 32;
      int bi = blk >> 3;
      int bj = blk & 7;
      int gi = rb * 8 + bi;
      int gj = ct * 8 + bj;
      int m0 = 2 * bi, m1 = 2 * bi + 1, n0 = 2 * bj, n1 = 2 * bj + 1;
      #pragma unroll
      for (int s = 0; s < 3; s++) {   // 0=lh, 1=hl, 2=hh
        float a = scw[s * 16 * TS + m0 * TS + n0];
        float b = scw[s * 16 * TS + m0 * TS + n1];
        float c = scw[s * 16 * TS + m1 * TS + n0];
        float d = scw[s * 16 * TS + m1 * TS + n1];
        int o1 = (s == 0) ? 0 : ((s == 1) ? 2 : 1);   // 15 / 75 / 45 deg
        int o2 = (s == 0) ? 5 : ((s == 1) ? 3 : 4);   // 165 / 105 / 135 deg
        size_t p1 = ((((size_t)bc * 6 + o1) * 128 + gi) * 128 + gj) * 2;
        yh_out[p1 + 0] = (a - d) * inv_sqrt2;
        yh_out[p1 + 1] = (b + c) * inv_sqrt2;
        size_t p2 = ((((size_t)bc * 6 + o2) * 128 + gi) * 128 + gj) * 2;
        yh_out[p2 + 0] = (a + d) * inv_sqrt2;
        yh_out[p2 + 1] = (b - c) * inv_sqrt2;
      }
    }
    __syncthreads();
  }
}

// ---------------------------------------------------------------------------
// Levels 2+: fused rowdfilt + coldfilt + q2c (stride-2, 10-tap, dual-phase).
// Decimated phase algebra (derived from xe[2::2]/xe[3::2] of the reference):
//   rowdfilt lo : even col -> fa=h0b @ 2tc-8;  odd col -> fb=h0a @ 2tc-9
//   rowdfilt hi : even col -> h1a @ 2tc-7;     odd col -> h1b @ 2tc-10 (swap)
//   coldfilt ns : even row -> fa @ 2or;        odd row -> fb @ 2or-1
//   coldfilt sw : even row -> fb @ 2or+1;      odd row -> fa @ 2or-2
// ---------------------------------------------------------------------------
__device__ __forceinline__ float rowdf10(const float* __restrict__ xr, int c, int ci0,
                                         const float* __restrict__ f) {
  float a = 0.0f;
  #pragma unroll
  for (int k = 0; k < 10; k++) a += f[k] * xr[reflect_idx(ci0 + 2 * k, c)];
  return a;
}

__device__ __forceinline__ float coldf10(const float* __restrict__ buf, int c2, int tc, int ri0,
                                         const float* __restrict__ f) {
  float a = 0.0f;
  #pragma unroll
  for (int k = 0; k < 10; k++) a += f[k] * buf[(ri0 + 2 * k) * c2 + tc];
  return a;
}

__device__ __forceinline__ float colf_ns(const float* buf, int c2, int tc, int orr,
                                         const float* fha, const float* fhb) {
  return ((orr & 1) == 0) ? coldf10(buf, c2, tc, 2 * orr, fha)
                          : coldf10(buf, c2, tc, 2 * orr - 1, fhb);
}

__device__ __forceinline__ float colf_sw(const float* buf, int c2, int tc, int orr,
                                         const float* fha, const float* fhb) {
  return ((orr & 1) == 0) ? coldf10(buf, c2, tc, 2 * orr + 1, fhb)
                          : coldf10(buf, c2, tc, 2 * orr - 2, fha);
}

__global__ __launch_bounds__(128) void dtcwt_l2plus(
    const float* __restrict__ x,
    const float* __restrict__ h0a, const float* __restrict__ h0b,
    const float* __restrict__ h1a, const float* __restrict__ h1b,
    float* __restrict__ ll_out, float* __restrict__ yh_out,
    int r, int c)
{
  extern __shared__ float smem[];
  const int c2 = c >> 1, c4 = c >> 2, r2 = r >> 1, r4 = r >> 2;
  float* xt = smem;               // 49*c
  float* lo = xt + 49 * c;        // 49*c2
  float* hi = lo + 49 * c2;       // 49*c2

  const int ob = blockIdx.x;      // block of 16 decimated output rows
  const int bc = blockIdx.y;
  const int t = threadIdx.x;

  const float* __restrict__ xin = x + (size_t)bc * r * c;
  const int csh = (c == 256) ? 8 : 7;
  const int base = 32 * ob - 8;   // input-row span [32ob-8, 32ob+40]

  // ---- Stage 1: global -> LDS (49 reflected input rows) ----
  for (int idx = t; idx < 49 * c; idx += 128) {
    int rr = idx >> csh;
    int cc = idx & (c - 1);
    int gr = reflect_idx(base + rr, r);
    xt[idx] = xin[(size_t)gr * c + cc];
  }
  __syncthreads();

  // ---- Stage 2: decimating row filters -> lo/hi (width c/2) ----
  if (t < c2) {
    const bool even = ((t & 1) == 0);
    for (int rr = 0; rr < 49; rr++) {
      const float* xr = xt + rr * c;
      float vLo, vHi;
      if (even) {
        vLo = rowdf10(xr, c, 2 * t - 8, h0b);   // ya phase, lowpass
        vHi = rowdf10(xr, c, 2 * t - 7, h1a);   // yb phase, highpass (swapped)
      } else {
        vLo = rowdf10(xr, c, 2 * t - 9, h0a);   // yb phase, lowpass
        vHi = rowdf10(xr, c, 2 * t - 10, h1b);  // ya phase, highpass (swapped)
      }
      lo[rr * c2 + t] = vLo;
      hi[rr * c2 + t] = vHi;
    }
  }
  __syncthreads();

  // ---- Stage 3a: LL (decimated both ways) ----
  if (t < c2) {
    for (int orr = 0; orr < 16; orr++) {
      int go = ob * 16 + orr;
      float v = colf_ns(lo, c2, t, orr, h0b, h0a);
      ll_out[((size_t)bc * r2 + go) * c2 + t] = v;
    }
  }

  // ---- Stage 3b: highpass column filters + q2c ----
  if (t < c4) {
    const float inv_sqrt2 = 0.70710678118654752440f;
    const int n0 = 2 * t, n1 = 2 * t + 1;
    for (int bi = 0; bi < 8; bi++) {
      int or0 = 2 * bi, or1 = 2 * bi + 1;
      int gi = ob * 8 + bi;
      int gj = t;
      // lh = coldfilt(lo, h1b, h1a, True)
      float la = colf_sw(lo, c2, n0, or0, h1b, h1a);
      float lb = colf_sw(lo, c2, n1, or0, h1b, h1a);
      float lc = colf_sw(lo, c2, n0, or1, h1b, h1a);
      float ld = colf_sw(lo, c2, n1, or1, h1b, h1a);
      // hl = coldfilt(hi, h0b, h0a, False)
      float pa = colf_ns(hi, c2, n0, or0, h0b, h0a);
      float pb = colf_ns(hi, c2, n1, or0, h0b, h0a);
      float pc = colf_ns(hi, c2, n0, or1, h0b, h0a);
      float pd = colf_ns(hi, c2, n1, or1, h0b, h0a);
      // hh = coldfilt(hi, h1b, h1a, True)
      float qa = colf_sw(hi, c2, n0, or0, h1b, h1a);
      float qb = colf_sw(hi, c2, n1, or0, h1b, h1a);
      float qc = colf_sw(hi, c2, n0, or1, h1b, h1a);
      float qd = colf_sw(hi, c2, n1, or1, h1b, h1a);

      size_t p1, p2;
      // lh -> orientations 0, 5
      p1 = ((((size_t)bc * 6 + 0) * r4 + gi) * c4 + gj) * 2;
      yh_out[p1 + 0] = (la - ld) * inv_sqrt2;  yh_out[p1 + 1] = (lb + lc) * inv_sqrt2;
      p2 = ((((size_t)bc * 6 + 5) * r4 + gi) * c4 + gj) * 2;
      yh_out[p2 + 0] = (la + ld) * inv_sqrt2;  yh_out[p2 + 1] = (lb - lc) * inv_sqrt2;
      // hh -> orientations 1, 4
      p1 = ((((size_t)bc * 6 + 1) * r4 + gi) * c4 + gj) * 2;
      yh_out[p1 + 0] = (qa - qd) * inv_sqrt2;  yh_out[p1 + 1] = (qb + qc) * inv_sqrt2;
      p2 = ((((size_t)bc * 6 + 4) * r4 + gi) * c4 + gj) * 2;
      yh_out[p2 + 0] = (qa + qd) * inv_sqrt2;  yh_out[p2 + 1] = (qb - qc) * inv_sqrt2;
      // hl -> orientations 2, 3
      p1 = ((((size_t)bc * 6 + 2) * r4 + gi) * c4 + gj) * 2;
      yh_out[p1 + 0] = (pa - pd) * inv_sqrt2;  yh_out[p1 + 1] = (pb + pc) * inv_sqrt2;
      p2 = ((((size_t)bc * 6 + 3) * r4 + gi) * c4 + gj) * 2;
      yh_out[p2 + 0] = (pa + pd) * inv_sqrt2;  yh_out[p2 + 1] = (pb - pc) * inv_sqrt2;
    }
  }
}

// ---------------------------------------------------------------------------
extern "C" void kernel_launch(void* const* d_in, const int* in_sizes, int n_in,
                              void* d_out, int out_size, void* d_ws, size_t ws_size,
                              hipStream_t stream) {
  (void)in_sizes; (void)n_in; (void)out_size; (void)ws_size;

  const float* x   = (const float*)d_in[0];
  const float* h0o = (const float*)d_in[1];
  const float* h1o = (const float*)d_in[2];
  const float* h0a = (const float*)d_in[3];
  const float* h0b = (const float*)d_in[4];
  const float* h1a = (const float*)d_in[5];
  const float* h1b = (const float*)d_in[6];

  float* out = (float*)d_out;
  float* ll_final = out;                         // (4,64,64,64)       1,048,576
  float* yh1 = out + (size_t)1048576;            // (4,64,6,128,128,2) 50,331,648
  float* yh2 = yh1 + (size_t)50331648;           // (4,64,6,64,64,2)   12,582,912
  float* yh3 = yh2 + (size_t)12582912;           // (4,64,6,32,32,2)    3,145,728

  float* ws1 = (float*)d_ws;                     // ll1 (4,64,256,256), 64 MiB
  float* ws2 = ws1 + (size_t)16777216;           // ll2 (4,64,128,128), 16 MiB

  // Level 1 (fused, WMMA column filter)
  size_t sh1 = (size_t)(3 * 22 * 256) * sizeof(float);
  dtcwt_l1<<<dim3(16, 256), 128, sh1, stream>>>(x, h0o, h1o, ws1, yh1);

  // Level 2: 256x256 -> ll2 128x128
  size_t sh2 = (size_t)(49 * 256 + 2 * 49 * 128) * sizeof(float);
  dtcwt_l2plus<<<dim3(8, 256), 128, sh2, stream>>>(ws1, h0a, h0b, h1a, h1b,
                                                   ws2, yh2, 256, 256);

  // Level 3: 128x128 -> ll_final 64x64
  size_t sh3 = (size_t)(49 * 128 + 2 * 49 * 64) * sizeof(float);
  dtcwt_l2plus<<<dim3(4, 256), 128, sh3, stream>>>(ws2, h0a, h0b, h1a, h1b,
                                                   ll_final, yh3, 128, 128);
}